// LightGlueMatchAssignmentLayer_39986145526095
// MI455X (gfx1250) — compile-verified
//
#include <hip/hip_runtime.h>
#include <math.h>

typedef float v2f __attribute__((ext_vector_type(2)));
typedef float v8f __attribute__((ext_vector_type(8)));

#define DIM    256
#define NPTS   2048
#define NPAIR  16
#define NBATCH 32
#define NOUT   2049
#define NEG_BIG (-3.4028234663852886e38f)

__device__ __forceinline__ float log_sigmoid_f(float x) {
    // stable: min(x,0) - log1p(exp(-|x|))
    return fminf(x, 0.0f) - log1pf(__expf(-fabsf(x)));
}

// ---------------------------------------------------------------------------
// 1) Projection: mdesc[r,c] = (sum_k desc[r,k]*Wp[c,k] + bp[c]) * 0.25
//    Wave tile: 32(M) x 64(N) = 2x4 tiles of 16x16, K-step 4 via
//    V_WMMA_F32_16X16X4_F32. Block = 8 waves -> 256(M) x 64(N).
// ---------------------------------------------------------------------------
__global__ void __launch_bounds__(256)
proj_kernel(const float* __restrict__ desc, const float* __restrict__ Wp,
            const float* __restrict__ bp, float* __restrict__ mdesc) {
    const int lane = threadIdx.x & 31;
    const int wave = threadIdx.x >> 5;
    const int lr = lane & 15;   // A: row M; B: col N; C: col N
    const int kh = lane >> 4;   // A/B: K-group; C: M+8 select
    const int m0 = blockIdx.x * 256 + wave * 32;
    const int n0 = blockIdx.y * 64;

    v8f acc[2][4] = {};
    const float* arow0 = desc + (size_t)(m0 + lr) * DIM + 2 * kh;
    const float* arow1 = arow0 + (size_t)16 * DIM;
    const float* brow  = Wp + (size_t)(n0 + lr) * DIM + 2 * kh;
    for (int k = 0; k < DIM; k += 4) {
        v2f a0 = *(const v2f*)(arow0 + k);
        v2f a1 = *(const v2f*)(arow1 + k);
#pragma unroll
        for (int j = 0; j < 4; ++j) {
            v2f b = *(const v2f*)(brow + j * 16 * DIM + k);
            acc[0][j] = __builtin_amdgcn_wmma_f32_16x16x4_f32(
                false, a0, false, b, (short)0, acc[0][j], false, false);
            acc[1][j] = __builtin_amdgcn_wmma_f32_16x16x4_f32(
                false, a1, false, b, (short)0, acc[1][j], false, false);
        }
    }
#pragma unroll
    for (int j = 0; j < 4; ++j) {
        const int n = n0 + j * 16 + lr;
        const float bias = bp[n];
#pragma unroll
        for (int t = 0; t < 2; ++t) {
#pragma unroll
            for (int r = 0; r < 8; ++r) {
                const int m = m0 + t * 16 + r + 8 * kh;
                mdesc[(size_t)m * DIM + n] = (acc[t][j][r] + bias) * 0.25f;
            }
        }
    }
}

// ---------------------------------------------------------------------------
// 2) Matchability: one wave per row; lsp = logsig(x), lsn = logsig(-x)
// ---------------------------------------------------------------------------
__global__ void __launch_bounds__(256)
match_kernel(const float* __restrict__ desc, const float* __restrict__ Wm,
             const float* __restrict__ bm, float* __restrict__ lsp,
             float* __restrict__ lsn) {
    const int lane = threadIdx.x & 31;
    const int wave = threadIdx.x >> 5;
    const int row = blockIdx.x * 8 + wave;
    const float* dr = desc + (size_t)row * DIM + lane * 8;
    const float* wr = Wm + lane * 8;
    float s = 0.0f;
#pragma unroll
    for (int j = 0; j < 8; ++j) s += dr[j] * wr[j];
#pragma unroll
    for (int off = 16; off > 0; off >>= 1) s += __shfl_down(s, off, 32);
    if (lane == 0) {
        const float x = s + bm[0];
        lsp[row] = log_sigmoid_f(x);
        lsn[row] = log_sigmoid_f(-x);
    }
}

// ---------------------------------------------------------------------------
// 3) Similarity per pair -> written into d_out inner region (stride NOUT)
//    Same 32x64 wave tile as proj_kernel.
// ---------------------------------------------------------------------------
__global__ void __launch_bounds__(256)
sim_kernel(const float* __restrict__ mdesc, const int* __restrict__ mask,
           float* __restrict__ out) {
    const int lane = threadIdx.x & 31;
    const int wave = threadIdx.x >> 5;
    const int lr = lane & 15;
    const int kh = lane >> 4;
    const int p = blockIdx.z;
    const int m0 = blockIdx.y * 256 + wave * 32;
    const int n0 = blockIdx.x * 64;
    const float* A  = mdesc + (size_t)(2 * p)     * NPTS * DIM;
    const float* Bm = mdesc + (size_t)(2 * p + 1) * NPTS * DIM;

    v8f acc[2][4] = {};
    const float* arow0 = A + (size_t)(m0 + lr) * DIM + 2 * kh;
    const float* arow1 = arow0 + (size_t)16 * DIM;
    const float* brow  = Bm + (size_t)(n0 + lr) * DIM + 2 * kh;
    for (int k = 0; k < DIM; k += 4) {
        v2f a0 = *(const v2f*)(arow0 + k);
        v2f a1 = *(const v2f*)(arow1 + k);
#pragma unroll
        for (int j = 0; j < 4; ++j) {
            v2f b = *(const v2f*)(brow + j * 16 * DIM + k);
            acc[0][j] = __builtin_amdgcn_wmma_f32_16x16x4_f32(
                false, a0, false, b, (short)0, acc[0][j], false, false);
            acc[1][j] = __builtin_amdgcn_wmma_f32_16x16x4_f32(
                false, a1, false, b, (short)0, acc[1][j], false, false);
        }
    }
    const int* mk0 = mask + (size_t)(2 * p) * NPTS;
    const int* mk1 = mask + (size_t)(2 * p + 1) * NPTS;
    float* outp = out + (size_t)p * NOUT * NOUT;
#pragma unroll
    for (int j = 0; j < 4; ++j) {
        const int n = n0 + j * 16 + lr;
        const int mk1v = mk1[n];
#pragma unroll
        for (int t = 0; t < 2; ++t) {
#pragma unroll
            for (int r = 0; r < 8; ++r) {
                const int m = m0 + t * 16 + r + 8 * kh;
                float v = acc[t][j][r];
                v = (mk0[m] && mk1v) ? v : NEG_BIG;
                outp[(size_t)m * NOUT + n] = v;
            }
        }
    }
}

// ---------------------------------------------------------------------------
// 4a) Row-wise LSE: one block per (p,m); 8 elements/thread kept in regs
// ---------------------------------------------------------------------------
__global__ void __launch_bounds__(256)
lse_row_kernel(const float* __restrict__ out, float* __restrict__ lse_row) {
    __shared__ float red[256];
    const int tid = threadIdx.x;
    const int row = blockIdx.x;          // p*2048 + m
    const int p = row >> 11;
    const int m = row & 2047;
    const float* rp = out + (size_t)p * NOUT * NOUT + (size_t)m * NOUT;
    float x[8];
    float mx = NEG_BIG;
#pragma unroll
    for (int j = 0; j < 8; ++j) { x[j] = rp[tid + j * 256]; mx = fmaxf(mx, x[j]); }
    red[tid] = mx; __syncthreads();
    for (int s = 128; s > 0; s >>= 1) {
        if (tid < s) red[tid] = fmaxf(red[tid], red[tid + s]);
        __syncthreads();
    }
    mx = red[0]; __syncthreads();
    float sum = 0.0f;
#pragma unroll
    for (int j = 0; j < 8; ++j) sum += __expf(x[j] - mx);
    red[tid] = sum; __syncthreads();
    for (int s = 128; s > 0; s >>= 1) {
        if (tid < s) red[tid] += red[tid + s];
        __syncthreads();
    }
    if (tid == 0) lse_row[row] = mx + __logf(red[0]);
}

// ---------------------------------------------------------------------------
// 4b) Column-wise online LSE, chunked over 256-row slabs (coalesced loads)
// ---------------------------------------------------------------------------
__global__ void __launch_bounds__(256)
lse_col_partial(const float* __restrict__ out, float* __restrict__ pmax,
                float* __restrict__ psum) {
    const int tid = threadIdx.x;
    const int p = blockIdx.z;
    const int mc = blockIdx.y;                  // 0..7 (slab of 256 rows)
    const int n = blockIdx.x * 256 + tid;
    const float* cp = out + (size_t)p * NOUT * NOUT + (size_t)(mc * 256) * NOUT + n;
    float M = NEG_BIG, S = 0.0f;
    for (int m = 0; m < 256; ++m) {
        const float xv = cp[(size_t)m * NOUT];
        if (xv > M) { S = S * __expf(M - xv) + 1.0f; M = xv; }
        else        { S += __expf(xv - M); }
    }
    const size_t idx = ((size_t)p * 8 + mc) * NPTS + n;
    pmax[idx] = M; psum[idx] = S;
}

__global__ void __launch_bounds__(256)
lse_col_merge(const float* __restrict__ pmax, const float* __restrict__ psum,
              float* __restrict__ lse_col) {
    const int idx = blockIdx.x * 256 + threadIdx.x;  // p*2048 + n
    const int p = idx >> 11;
    const int n = idx & 2047;
    float M = NEG_BIG;
#pragma unroll
    for (int c = 0; c < 8; ++c)
        M = fmaxf(M, pmax[((size_t)p * 8 + c) * NPTS + n]);
    float S = 0.0f;
#pragma unroll
    for (int c = 0; c < 8; ++c) {
        const size_t i = ((size_t)p * 8 + c) * NPTS + n;
        S += psum[i] * __expf(pmax[i] - M);
    }
    lse_col[idx] = M + __logf(S);
}

// ---------------------------------------------------------------------------
// 5) Final in-place assembly of (P, 2049, 2049)
// ---------------------------------------------------------------------------
__global__ void __launch_bounds__(256)
final_kernel(float* __restrict__ out, const float* __restrict__ lse_row,
             const float* __restrict__ lse_col, const float* __restrict__ lsp,
             const float* __restrict__ lsn) {
    const size_t per = (size_t)NOUT * NOUT;
    const size_t e = (size_t)blockIdx.x * 256 + threadIdx.x;
    if (e >= per) return;
    const int p = blockIdx.y;
    const int i = (int)(e / NOUT);
    const int j = (int)(e % NOUT);
    float* op = out + (size_t)p * per;
    float v;
    if (i < NPTS && j < NPTS) {
        const float s = op[e];
        v = 2.0f * s - lse_row[p * NPTS + i] - lse_col[p * NPTS + j]
            + lsp[(2 * p) * NPTS + i] + lsp[(2 * p + 1) * NPTS + j];
    } else if (i < NPTS) {           // j == N : logsig(-m0)
        v = lsn[(2 * p) * NPTS + i];
    } else if (j < NPTS) {           // i == N : logsig(-m1)
        v = lsn[(2 * p + 1) * NPTS + j];
    } else {
        v = 0.0f;
    }
    op[e] = v;
}

// ---------------------------------------------------------------------------
extern "C" void kernel_launch(void* const* d_in, const int* in_sizes, int n_in,
                              void* d_out, int out_size, void* d_ws, size_t ws_size,
                              hipStream_t stream) {
    (void)in_sizes; (void)n_in; (void)out_size; (void)ws_size;
    const float* desc = (const float*)d_in[0];
    const int*   mask = (const int*)d_in[1];
    const float* Wp   = (const float*)d_in[2];
    const float* bp   = (const float*)d_in[3];
    const float* Wm   = (const float*)d_in[4];
    const float* bm   = (const float*)d_in[5];
    float* out = (float*)d_out;

    char* ws = (char*)d_ws;
    float* mdesc   = (float*)ws; ws += (size_t)NBATCH * NPTS * DIM * sizeof(float); // 64 MB
    float* lsp     = (float*)ws; ws += (size_t)NBATCH * NPTS * sizeof(float);
    float* lsn     = (float*)ws; ws += (size_t)NBATCH * NPTS * sizeof(float);
    float* lse_row = (float*)ws; ws += (size_t)NPAIR * NPTS * sizeof(float);
    float* lse_col = (float*)ws; ws += (size_t)NPAIR * NPTS * sizeof(float);
    float* pmax    = (float*)ws; ws += (size_t)NPAIR * 8 * NPTS * sizeof(float);
    float* psum    = (float*)ws; ws += (size_t)NPAIR * 8 * NPTS * sizeof(float);

    // 1) projection GEMM (WMMA f32), 32x64 wave tiles
    proj_kernel<<<dim3((NBATCH * NPTS) / 256, DIM / 64), 256, 0, stream>>>(desc, Wp, bp, mdesc);
    // 2) matchability GEMV + log-sigmoids
    match_kernel<<<(NBATCH * NPTS) / 8, 256, 0, stream>>>(desc, Wm, bm, lsp, lsn);
    // 3) similarity GEMM (WMMA f32) -> d_out inner region
    sim_kernel<<<dim3(NPTS / 64, NPTS / 256, NPAIR), 256, 0, stream>>>(mdesc, mask, out);
    // 4) bidirectional LSE
    lse_row_kernel<<<NPAIR * NPTS, 256, 0, stream>>>(out, lse_row);
    lse_col_partial<<<dim3(NPTS / 256, 8, NPAIR), 256, 0, stream>>>(out, pmax, psum);
    lse_col_merge<<<(NPAIR * NPTS) / 256, 256, 0, stream>>>(pmax, psum, lse_col);
    // 5) final assembly (in place) + borders
    const int nblk = (int)(((size_t)NOUT * NOUT + 255) / 256);
    final_kernel<<<dim3(nblk, NPAIR), 256, 0, stream>>>(out, lse_row, lse_col, lsp, lsn);
}